// ChunkedDFT_57664230916161
// MI455X (gfx1250) — compile-verified
//
#include <hip/hip_runtime.h>

// Chunked non-uniform DFT for MI455X (gfx1250, wave32).
// vis[b,k] = sum_p exp(-2pi*i*(u_k*l_p + v_k*m_p + w_k*(n_p-1))) * sky[b,p]
//
// Strategy:
//  * t[k,p] = -(u*l + v*m + w*nm1) computed as a 16x16 tile with one
//    V_WMMA_F32_16X16X4_F32 (A = negated u/v/w per k, loop-invariant;
//    B = l/m/nm1 pre-packed in LDS in B-operand layout).
//  * sin(2pi*t)/cos(2pi*t) via hardware v_sin/v_cos (revolution input).
//  * complex MAC accumulated in the WMMA C/D register layout (lane = p,
//    VGPR = k row), cross-lane p-reduction done once at the end.

#define P_PIX 16384
#define V_VIS 16384
#define CP    1024                 // pixels staged per LDS chunk
#define THREADS 256
#define WAVES (THREADS / 32)       // 8 waves
#define K_PER_BLOCK (WAVES * 16)   // 128 visibilities per block

typedef __attribute__((ext_vector_type(2))) float v2f;
typedef __attribute__((ext_vector_type(8))) float v8f;

__global__ __launch_bounds__(THREADS)
void ChunkedDFT_wmma_kernel(const float* __restrict__ sky_r,
                            const float* __restrict__ sky_i,
                            const float* __restrict__ lc,
                            const float* __restrict__ mc,
                            const float* __restrict__ nc,
                            const float* __restrict__ uc,
                            const float* __restrict__ vc,
                            const float* __restrict__ wc,
                            float* __restrict__ out)
{
    // B-operand tiles: per 16-pixel tile, 32 lanes x 2 dwords, packed so that
    // lane L reads a contiguous float2 {b0,b1}:
    //   lanes 0-15 : {l[p0+L],      m[p0+L]}      (K rows 0,1)
    //   lanes 16-31: {nm1[p0+L-16], 0.0f}         (K rows 2,3)
    __shared__ float lds_b[(CP / 16) * 64];   // 16 KB
    __shared__ float lds_s[CP * 4];           // 16 KB: {sr0,si0,sr1,si1} per p

    const int lane = threadIdx.x & 31;
    const int wave = threadIdx.x >> 5;
    const int k0   = blockIdx.x * K_PER_BLOCK + wave * 16;
    const int kk   = k0 + (lane & 15);

    // A operand (16x4 f32 layout): VGPR0 = K0 (lanes 0-15) / K2 (lanes 16-31),
    // VGPR1 = K1 / K3. Pre-negate to fold the "-2pi" sign; f32 WMMA has no A-neg.
    v2f amat;
    amat.x = (lane < 16) ? -uc[kk] : -wc[kk];
    amat.y = (lane < 16) ? -vc[kk] : 0.0f;

    // Accumulators live in the WMMA C/D layout:
    //   VGPR g, lanes 0-15  -> (k = k0+g,   p = lane)
    //   VGPR g, lanes 16-31 -> (k = k0+8+g, p = lane-16)
    float aR0[8], aI0[8], aR1[8], aI1[8];
#pragma unroll
    for (int g = 0; g < 8; ++g) { aR0[g] = aI0[g] = aR1[g] = aI1[g] = 0.0f; }

    for (int pc = 0; pc < P_PIX; pc += CP) {
        __syncthreads();   // previous chunk fully consumed
        // ---- stage chunk into LDS (4 pixels per thread) ----
#pragma unroll
        for (int i = 0; i < CP / THREADS; ++i) {
            const int j = threadIdx.x + i * THREADS;   // 0..CP-1
            const int p = pc + j;
            const float l   = lc[p];
            const float m   = mc[p];
            const float nm1 = nc[p] - 1.0f;
            float* bp = &lds_b[(j >> 4) * 64];
            const int e = j & 15;
            v2f lm; lm.x = l;   lm.y = m;
            v2f nz; nz.x = nm1; nz.y = 0.0f;
            ((v2f*)bp)[e]      = lm;   // lane slot e      (K rows 0,1)
            ((v2f*)bp)[e + 16] = nz;   // lane slot e + 16 (K rows 2,3)
            float4 s;
            s.x = sky_r[p];           s.y = sky_i[p];
            s.z = sky_r[P_PIX + p];   s.w = sky_i[P_PIX + p];
            ((float4*)lds_s)[j] = s;
        }
        __syncthreads();

        // ---- compute: every wave sweeps all tiles of the chunk ----
        for (int t = 0; t < CP / 16; ++t) {
            const v2f b = ((const v2f*)&lds_b[t * 64])[lane];
            v8f cz = {};
            // t[k,p] in revolutions: one matrix op replaces 3 FMAs/element.
            v8f ph = __builtin_amdgcn_wmma_f32_16x16x4_f32(
                false, amat, false, b, (short)0, cz, false, false);
            const float4 s = ((const float4*)lds_s)[t * 16 + (lane & 15)];
#pragma unroll
            for (int g = 0; g < 8; ++g) {
                const float sn = __builtin_amdgcn_sinf(ph[g]);  // sin(2pi*t)
                const float cs = __builtin_amdgcn_cosf(ph[g]);  // cos(2pi*t)
                aR0[g] = fmaf(cs, s.x, aR0[g]); aR0[g] = fmaf(-sn, s.y, aR0[g]);
                aI0[g] = fmaf(cs, s.y, aI0[g]); aI0[g] = fmaf( sn, s.x, aI0[g]);
                aR1[g] = fmaf(cs, s.z, aR1[g]); aR1[g] = fmaf(-sn, s.w, aR1[g]);
                aI1[g] = fmaf(cs, s.w, aI1[g]); aI1[g] = fmaf( sn, s.z, aI1[g]);
            }
        }
    }

    // ---- p-reduction across each 16-lane half (xor masks stay in-half) ----
#pragma unroll
    for (int g = 0; g < 8; ++g) {
        for (int msk = 1; msk <= 8; msk <<= 1) {
            aR0[g] += __shfl_xor(aR0[g], msk, 32);
            aI0[g] += __shfl_xor(aI0[g], msk, 32);
            aR1[g] += __shfl_xor(aR1[g], msk, 32);
            aI1[g] += __shfl_xor(aI1[g], msk, 32);
        }
    }

    // Output: complex64 [B, V] flattened as float pairs.
    if ((lane & 15) == 0) {
        const int kb = k0 + ((lane >> 4) << 3);   // lane 0 -> k0, lane 16 -> k0+8
#pragma unroll
        for (int g = 0; g < 8; ++g) {
            const int k = kb + g;
            out[2 * k + 0]             = aR0[g];
            out[2 * k + 1]             = aI0[g];
            out[2 * (V_VIS + k) + 0]   = aR1[g];
            out[2 * (V_VIS + k) + 1]   = aI1[g];
        }
    }
}

extern "C" void kernel_launch(void* const* d_in, const int* in_sizes, int n_in,
                              void* d_out, int out_size, void* d_ws, size_t ws_size,
                              hipStream_t stream) {
    const float* sky_r = (const float*)d_in[0];
    const float* sky_i = (const float*)d_in[1];
    const float* lc    = (const float*)d_in[2];
    const float* mc    = (const float*)d_in[3];
    const float* nc    = (const float*)d_in[4];
    const float* uc    = (const float*)d_in[5];
    const float* vc    = (const float*)d_in[6];
    const float* wc    = (const float*)d_in[7];
    float* out = (float*)d_out;

    dim3 grid(V_VIS / K_PER_BLOCK);   // 128 blocks
    dim3 block(THREADS);              // 8 waves (wave32)
    ChunkedDFT_wmma_kernel<<<grid, block, 0, stream>>>(
        sky_r, sky_i, lc, mc, nc, uc, vc, wc, out);
}